// Net_90288802496765
// MI455X (gfx1250) — compile-verified
//
#include <hip/hip_runtime.h>
#include <math.h>

#define N_NODES 8192
#define N_EDGES 131072
#define F_HID   128
#define N_C     16

typedef __attribute__((ext_vector_type(2))) float v2f;
typedef __attribute__((ext_vector_type(8))) float v8f;

// ---- workspace layout (float offsets) ----
#define WS_AGG   0u          // 8192*128 ; becomes h (relu'd GCN out) in-place
#define WS_H0    1048576u    // 8192*128
#define WS_S     2097152u    // 8192*16  (aligned copy of s)
#define WS_DEGE  2228224u    // 8192  raw in-degree (no self loop)
#define WS_DINV  2236416u    // 8192
#define WS_OUTA  2244608u    // 16*128  s^T h accumulator
#define WS_SS    2246656u    // 16*16   s^T s accumulator
#define WS_CA    2246912u    // 16
#define WS_CS    2246928u    // 16
#define WS_TR    2246944u    // 1   trace(s^T A s)
#define WS_Z2_BEG WS_DEGE
#define WS_Z2_CNT (WS_TR + 1u - WS_DEGE)

__global__ void zero_kernel(float* __restrict__ p, unsigned n) {
    unsigned i = blockIdx.x * blockDim.x + threadIdx.x;
    if (i < n) p[i] = 0.0f;
}

// h0 = x @ W1   (x: [N,4], W1: [4,128])
__global__ void h0_kernel(const float* __restrict__ x, const float* __restrict__ W1,
                          float* __restrict__ h0) {
    unsigned i = blockIdx.x * blockDim.x + threadIdx.x;   // n*128+f
    if (i >= N_NODES * F_HID) return;
    unsigned n = i >> 7, f = i & 127u;
    const float* xr = x + n * 4u;
    h0[i] = xr[0]*W1[f] + xr[1]*W1[128u+f] + xr[2]*W1[256u+f] + xr[3]*W1[384u+f];
}

__global__ void deg_kernel(const int* __restrict__ dst, float* __restrict__ degE) {
    unsigned e = blockIdx.x * blockDim.x + threadIdx.x;
    if (e < N_EDGES) atomicAdd(&degE[dst[e]], 1.0f);
}

__global__ void dinv_kernel(const float* __restrict__ degE, float* __restrict__ dinv) {
    unsigned n = blockIdx.x * blockDim.x + threadIdx.x;
    if (n < N_NODES) dinv[n] = rsqrtf(degE[n] + 1.0f);   // +1 self loop
}

// one wave per edge: agg[dst] += h0[src] * dinv[src]*dinv[dst]
__global__ void __launch_bounds__(256) scatter_kernel(const int* __restrict__ src,
        const int* __restrict__ dst, const float* __restrict__ h0,
        const float* __restrict__ dinv, float* __restrict__ agg) {
    unsigned e    = (blockIdx.x * blockDim.x + threadIdx.x) >> 5;
    unsigned lane = threadIdx.x & 31u;
    if (e >= N_EDGES) return;
    int s = src[e], d = dst[e];
    float nrm = dinv[s] * dinv[d];
    float4 v = ((const float4*)(h0 + (size_t)s * F_HID))[lane];
    float* a = agg + (size_t)d * F_HID + lane * 4u;
    atomicAdd(a + 0, v.x * nrm);
    atomicAdd(a + 1, v.y * nrm);
    atomicAdd(a + 2, v.z * nrm);
    atomicAdd(a + 3, v.w * nrm);
}

// h = relu(agg + h0*dinv^2 + b1), written over agg
__global__ void hfinal_kernel(float* __restrict__ agg, const float* __restrict__ h0,
                              const float* __restrict__ dinv, const float* __restrict__ b1) {
    unsigned i = blockIdx.x * blockDim.x + threadIdx.x;
    if (i >= N_NODES * F_HID) return;
    unsigned n = i >> 7, f = i & 127u;
    float di = dinv[n];
    float v = agg[i] + h0[i] * di * di + b1[f];
    agg[i] = v > 0.0f ? v : 0.0f;
}

// per-node: logits = h@Wp + bp (Wp in LDS), softmax -> s; accumulate cluster_size, ca
__global__ void __launch_bounds__(256) softmax_kernel(const float* __restrict__ h,
        const float* __restrict__ Wp, const float* __restrict__ bp,
        const float* __restrict__ degE, float* __restrict__ s_ws,
        float* __restrict__ s_out, float* __restrict__ ca, float* __restrict__ cs) {
    __shared__ float WpL[F_HID * N_C];
    __shared__ float caL[N_C], csL[N_C];
    unsigned tid = threadIdx.x;
    for (unsigned i = tid; i < F_HID * N_C; i += 256u) WpL[i] = Wp[i];
    if (tid < N_C) { caL[tid] = 0.0f; csL[tid] = 0.0f; }
    __syncthreads();

    unsigned n = blockIdx.x * 256u + tid;   // grid sized exactly N_NODES
    float lg[N_C];
    #pragma unroll
    for (int c = 0; c < N_C; ++c) lg[c] = bp[c];
    const float* hr = h + (size_t)n * F_HID;
    for (int f = 0; f < F_HID; ++f) {
        float hv = hr[f];
        #pragma unroll
        for (int c = 0; c < N_C; ++c) lg[c] += hv * WpL[f * N_C + c];
    }
    float mx = lg[0];
    #pragma unroll
    for (int c = 1; c < N_C; ++c) mx = fmaxf(mx, lg[c]);
    float sum = 0.0f;
    #pragma unroll
    for (int c = 0; c < N_C; ++c) { lg[c] = expf(lg[c] - mx); sum += lg[c]; }
    float inv = 1.0f / sum;
    float dg  = degE[n];
    #pragma unroll
    for (int c = 0; c < N_C; ++c) {
        float sv = lg[c] * inv;
        s_ws[n * N_C + c]  = sv;
        s_out[n * N_C + c] = sv;
        atomicAdd(&csL[c], sv);
        atomicAdd(&caL[c], sv * dg);
    }
    __syncthreads();
    if (tid < N_C) { atomicAdd(&cs[tid], csL[tid]); atomicAdd(&ca[tid], caL[tid]); }
}

// out = s^T h (16x128) and ss = s^T s (16x16) via V_WMMA_F32_16X16X4_F32, split-K
__global__ void __launch_bounds__(256) sth_wmma_kernel(const float* __restrict__ s,
        const float* __restrict__ h, float* __restrict__ outA, float* __restrict__ ssA) {
    unsigned wave = (blockIdx.x * blockDim.x + threadIdx.x) >> 5;   // 0..127
    unsigned lane = threadIdx.x & 31u;
    unsigned m  = lane & 15u;           // row of A (cluster) / col of B
    unsigned k2 = (lane >> 4) << 1;     // K offset pair for this half-wave

    v8f zero8 = {0.f,0.f,0.f,0.f,0.f,0.f,0.f,0.f};
    v8f cacc[8];
    #pragma unroll
    for (int t = 0; t < 8; ++t) cacc[t] = zero8;
    v8f css = zero8;

    unsigned nbase = wave * 64u;                    // 64 nodes per wave
    for (unsigned kb = nbase; kb < nbase + 64u; kb += 4u) {
        unsigned r0 = kb + k2, r1 = r0 + 1u;
        v2f a;                                      // A[c][k] = s[k][c]
        a.x = s[r0 * N_C + m];
        a.y = s[r1 * N_C + m];
        // ss: B[k][c'] = s[k][c']  -> identical per-lane gather as A
        css = __builtin_amdgcn_wmma_f32_16x16x4_f32(false, a, false, a,
                                                    (short)0, css, false, false);
        #pragma unroll
        for (int t = 0; t < 8; ++t) {               // B[k][f] = h[k][t*16+f]
            v2f b;
            b.x = h[(size_t)r0 * F_HID + t * 16u + m];
            b.y = h[(size_t)r1 * F_HID + t * 16u + m];
            cacc[t] = __builtin_amdgcn_wmma_f32_16x16x4_f32(false, a, false, b,
                                                            (short)0, cacc[t], false, false);
        }
    }
    unsigned Mb = (lane >> 4) << 3;                 // C/D layout: M = v + 8*(lane>=16)
    #pragma unroll
    for (int v = 0; v < 8; ++v) {
        unsigned M = v + Mb;
        atomicAdd(&ssA[M * N_C + m], css[v]);
        #pragma unroll
        for (int t = 0; t < 8; ++t)
            atomicAdd(&outA[M * F_HID + t * 16u + m], cacc[t][v]);
    }
}

// trace(s^T A s) = sum_e dot(s[src_e], s[dst_e])
__global__ void __launch_bounds__(256) trace_kernel(const int* __restrict__ src,
        const int* __restrict__ dst, const float* __restrict__ s, float* __restrict__ tr) {
    __shared__ float acc;
    if (threadIdx.x == 0) acc = 0.0f;
    __syncthreads();
    unsigned e = blockIdx.x * blockDim.x + threadIdx.x;
    float d = 0.0f;
    if (e < N_EDGES) {
        const float4* a = (const float4*)(s + (size_t)src[e] * N_C);
        const float4* b = (const float4*)(s + (size_t)dst[e] * N_C);
        #pragma unroll
        for (int i = 0; i < 4; ++i) {
            float4 av = a[i], bv = b[i];
            d += av.x*bv.x + av.y*bv.y + av.z*bv.z + av.w*bv.w;
        }
    }
    for (int o = 16; o; o >>= 1) d += __shfl_xor(d, o);
    if ((threadIdx.x & 31u) == 0) atomicAdd(&acc, d);
    __syncthreads();
    if (threadIdx.x == 0) atomicAdd(tr, acc);
}

// SELU + row log-softmax of out (16 waves, 1 row each) + scalar loss assembly
__global__ void __launch_bounds__(512) final_kernel(const float* __restrict__ ws,
                                                    float* __restrict__ out) {
    const float SC = 1.0507009873554805f, AL = 1.6732632423543772f;
    unsigned w = threadIdx.x >> 5, lane = threadIdx.x & 31u;   // w = row c, 0..15
    const float* outA = ws + WS_OUTA;
    float v[4];
    float mx = -3.0e38f;
    #pragma unroll
    for (int i = 0; i < 4; ++i) {
        float t = outA[w * F_HID + lane + 32u * i];
        t = t > 0.0f ? SC * t : SC * AL * (expf(t) - 1.0f);    // selu
        v[i] = t; mx = fmaxf(mx, t);
    }
    for (int o = 16; o; o >>= 1) mx = fmaxf(mx, __shfl_xor(mx, o));
    float se = 0.0f;
    #pragma unroll
    for (int i = 0; i < 4; ++i) se += expf(v[i] - mx);
    for (int o = 16; o; o >>= 1) se += __shfl_xor(se, o);
    float lse = logf(se);
    #pragma unroll
    for (int i = 0; i < 4; ++i)
        out[w * F_HID + lane + 32u * i] = v[i] - mx - lse;

    if (threadIdx.x == 0) {
        const float* ss = ws + WS_SS;
        const float* ca = ws + WS_CA;
        const float* cs = ws + WS_CS;
        float m   = (float)N_EDGES * 0.5f;      // sum(deg)/2 == E/2
        float trA = ws[WS_TR];
        float caSq = 0.0f;
        for (int c = 0; c < N_C; ++c) caSq += ca[c] * ca[c];
        float spectral = -(trA - caSq / (2.0f * m)) / (2.0f * m);
        float ssSq = 0.0f;
        for (int i = 0; i < N_C * N_C; ++i) ssSq += ss[i] * ss[i];
        float fro = sqrtf(ssSq);
        float dsum = 0.0f;
        for (int i = 0; i < N_C; ++i)
            for (int j = 0; j < N_C; ++j) {
                float dd = ss[i * N_C + j] / fro - ((i == j) ? 0.25f : 0.0f); // 1/sqrt(16)
                dsum += dd * dd;
            }
        float ortho = sqrtf(dsum);
        float csq = 0.0f;
        for (int c = 0; c < N_C; ++c) csq += cs[c] * cs[c];
        float cluster = sqrtf(csq) / (float)N_NODES * 4.0f - 1.0f;  // *sqrt(16)
        out[2048] = spectral + ortho + cluster;
    }
}

extern "C" void kernel_launch(void* const* d_in, const int* in_sizes, int n_in,
                              void* d_out, int out_size, void* d_ws, size_t ws_size,
                              hipStream_t stream) {
    const float* x  = (const float*)d_in[0];
    const float* W1 = (const float*)d_in[1];
    const float* b1 = (const float*)d_in[2];
    const float* Wp = (const float*)d_in[3];
    const float* bp = (const float*)d_in[4];
    const int*   ei = (const int*)d_in[5];
    const int* src = ei;
    const int* dst = ei + N_EDGES;
    float* out = (float*)d_out;
    float* ws  = (float*)d_ws;

    zero_kernel<<<(N_NODES * F_HID + 255) / 256, 256, 0, stream>>>(ws + WS_AGG, N_NODES * F_HID);
    zero_kernel<<<(WS_Z2_CNT + 255) / 256, 256, 0, stream>>>(ws + WS_Z2_BEG, WS_Z2_CNT);
    h0_kernel<<<(N_NODES * F_HID + 255) / 256, 256, 0, stream>>>(x, W1, ws + WS_H0);
    deg_kernel<<<N_EDGES / 256, 256, 0, stream>>>(dst, ws + WS_DEGE);
    dinv_kernel<<<N_NODES / 256, 256, 0, stream>>>(ws + WS_DEGE, ws + WS_DINV);
    scatter_kernel<<<N_EDGES / 8, 256, 0, stream>>>(src, dst, ws + WS_H0, ws + WS_DINV, ws + WS_AGG);
    hfinal_kernel<<<(N_NODES * F_HID + 255) / 256, 256, 0, stream>>>(ws + WS_AGG, ws + WS_H0,
                                                                     ws + WS_DINV, b1);
    softmax_kernel<<<N_NODES / 256, 256, 0, stream>>>(ws + WS_AGG, Wp, bp, ws + WS_DEGE,
                                                      ws + WS_S, out + 2049, ws + WS_CA, ws + WS_CS);
    sth_wmma_kernel<<<16, 256, 0, stream>>>(ws + WS_S, ws + WS_AGG, ws + WS_OUTA, ws + WS_SS);
    trace_kernel<<<N_EDGES / 256, 256, 0, stream>>>(src, dst, ws + WS_S, ws + WS_TR);
    final_kernel<<<1, 512, 0, stream>>>(ws, out);
}